// ErrorMinimizationLoss_70497593196920
// MI455X (gfx1250) — compile-verified
//
#include <hip/hip_runtime.h>
#include <hip/hip_bf16.h>

// ---------------------------------------------------------------------------
// Compile-time reconstruction of _build_tables() from the reference.
// Codon list order == base-4 index: idx = 4*d(c0) + 16*d(c1) + d(c2),
// with U=0, C=1, A=2, G=3 (verified against the Python dict order).
// ---------------------------------------------------------------------------

struct Tables { float loss[64]; int cnt[64]; };

constexpr double cabsd(double x) { return x < 0.0 ? -x : x; }

constexpr int aaIdx(char a) {
    const char* o = "ARNDCQEGHILKMFPSTWYV";
    for (int i = 0; i < 20; ++i) if (o[i] == a) return i;
    return 0;
}

constexpr int ntDigit(char c) {
    return c == 'U' ? 0 : (c == 'C' ? 1 : (c == 'A' ? 2 : 3));
}

constexpr Tables build_tables() {
    // AA_PROPS in dict order A R N D C Q E G H I L K M F P S T W Y V
    const double props[20][4] = {
        { 1.8,  88.6,  0.0,  0.0}, {-4.5, 173.4, 52.0,  1.0},
        {-3.5, 114.1,  3.38, 0.0}, {-3.5, 111.1, 49.7, -1.0},
        { 2.5, 108.5,  1.48, 0.0}, {-3.5, 143.8,  3.53, 0.0},
        {-3.5, 138.4, 49.9, -1.0}, {-0.4,  60.1,  0.0,  0.0},
        {-3.2, 153.2, 51.6,  0.5}, { 4.5, 166.7,  0.13, 0.0},
        { 3.8, 166.7,  0.13, 0.0}, {-3.9, 168.6, 49.5,  1.0},
        { 1.9, 162.9,  1.43, 0.0}, { 2.8, 189.9,  0.35, 0.0},
        {-1.6, 112.7,  1.58, 0.0}, {-0.8,  89.0,  1.67, 0.0},
        {-0.7, 116.1,  1.66, 0.0}, {-0.9, 227.8,  2.1,  0.0},
        {-1.3, 193.6,  1.61, 0.0}, { 4.2, 140.0,  0.13, 0.0}
    };
    const double w[4] = {0.4, 0.3, 0.2, 0.1};

    double maxes[4] = {0, 0, 0, 0};
    for (int i = 0; i < 20; ++i)
        for (int p = 0; p < 4; ++p) {
            double a = cabsd(props[i][p]);
            if (a > maxes[p]) maxes[p] = a;
        }
    double scale[4] = {0, 0, 0, 0};
    for (int p = 0; p < 4; ++p)
        scale[p] = maxes[p] > 0.0 ? w[p] / maxes[p] : 0.0;

    double Dm[20][20] = {};
    for (int i = 0; i < 20; ++i)
        for (int j = 0; j < 20; ++j) {
            double s = 0.0;
            for (int p = 0; p < 4; ++p)
                s += cabsd(props[i][p] - props[j][p]) * scale[p];
            Dm[i][j] = s;
        }

    // CODON_TABLE (dict order); placed into aaOf[] by base-4 index.
    struct CE { const char* c; char a; };
    const CE tab[64] = {
        {"UUU",'F'},{"UUC",'F'},{"UUA",'L'},{"UUG",'L'},
        {"CUU",'L'},{"CUC",'L'},{"CUA",'L'},{"CUG",'L'},
        {"AUU",'I'},{"AUC",'I'},{"AUA",'I'},{"AUG",'M'},
        {"GUU",'V'},{"GUC",'V'},{"GUA",'V'},{"GUG",'V'},
        {"UCU",'S'},{"UCC",'S'},{"UCA",'S'},{"UCG",'S'},
        {"CCU",'P'},{"CCC",'P'},{"CCA",'P'},{"CCG",'P'},
        {"ACU",'T'},{"ACC",'T'},{"ACA",'T'},{"ACG",'T'},
        {"GCU",'A'},{"GCC",'A'},{"GCA",'A'},{"GCG",'A'},
        {"UAU",'Y'},{"UAC",'Y'},{"UAA",'*'},{"UAG",'*'},
        {"CAU",'H'},{"CAC",'H'},{"CAA",'Q'},{"CAG",'Q'},
        {"AAU",'N'},{"AAC",'N'},{"AAA",'K'},{"AAG",'K'},
        {"GAU",'D'},{"GAC",'D'},{"GAA",'E'},{"GAG",'E'},
        {"UGU",'C'},{"UGC",'C'},{"UGA",'*'},{"UGG",'W'},
        {"CGU",'R'},{"CGC",'R'},{"CGA",'R'},{"CGG",'R'},
        {"AGU",'S'},{"AGC",'S'},{"AGA",'R'},{"AGG",'R'},
        {"GGU",'G'},{"GGC",'G'},{"GGA",'G'},{"GGG",'G'}
    };
    char aaOf[64] = {};
    for (int e = 0; e < 64; ++e) {
        int i0 = ntDigit(tab[e].c[0]);
        int i1 = ntDigit(tab[e].c[1]);
        int i2 = ntDigit(tab[e].c[2]);
        aaOf[4 * i0 + 16 * i1 + i2] = tab[e].a;
    }

    Tables T{};
    double lossD[64] = {};
    for (int ci = 0; ci < 64; ++ci) { T.loss[ci] = 0.0f; T.cnt[ci] = 0; }

    for (int ci = 0; ci < 64; ++ci) {
        char aa = aaOf[ci];
        if (aa == '*') continue;
        int dig[3] = { (ci >> 2) & 3, (ci >> 4) & 3, ci & 3 };  // c0,c1,c2 digits
        for (int pos = 0; pos < 3; ++pos) {
            for (int nt = 0; nt < 4; ++nt) {       // 'UCAG' order == digit order
                if (nt == dig[pos]) continue;
                int nd0 = dig[0], nd1 = dig[1], nd2 = dig[2];
                if (pos == 0) nd0 = nt; else if (pos == 1) nd1 = nt; else nd2 = nt;
                int ni = 4 * nd0 + 16 * nd1 + nd2;
                char naa = aaOf[ni];
                if (naa == '*') continue;
                int d = ci - ni; if (d < 0) d = -d;
                int v = 0, t = d;
                while (t > 0 && t % 3 == 0) { ++v; t /= 3; }
                double padic = (d > 0) ? 1.0 : 0.0;
                for (int q = 0; q < v; ++q) padic /= 3.0;
                double dd = Dm[aaIdx(aa)][aaIdx(naa)] - padic;
                lossD[ci] += dd * dd;
                T.cnt[ci] += 1;
            }
        }
    }
    for (int ci = 0; ci < 64; ++ci) T.loss[ci] = (float)lossD[ci];
    return T;
}

__constant__ Tables kTab = build_tables();

// ---------------------------------------------------------------------------
// Kernel 1: deterministic 64-bin histogram of codon indices.
// Per-block LDS histogram (ds_add), per-block partials to d_ws (no atomics
// across blocks -> fully deterministic).
// ---------------------------------------------------------------------------
#define NBLK 128
#define NTHR 256

__global__ void hist_kernel(const int* __restrict__ idx, int n,
                            unsigned int* __restrict__ partial) {
    __shared__ unsigned int lh[64];
    for (int i = threadIdx.x; i < 64; i += blockDim.x) lh[i] = 0u;
    __syncthreads();

    int stride = gridDim.x * blockDim.x;
    for (int i = blockIdx.x * blockDim.x + threadIdx.x; i < n; i += stride) {
        __builtin_prefetch(&idx[i + stride], 0, 0);   // global_prefetch_b8
        int v = idx[i];
        if ((unsigned)v < 64u) atomicAdd(&lh[v], 1u); // ds_add_u32
    }
    __syncthreads();
    for (int c = threadIdx.x; c < 64; c += blockDim.x)
        partial[blockIdx.x * 64 + c] = lh[c];
}

// ---------------------------------------------------------------------------
// Kernel 2: one wave. Reduce per-block partials, then compute the two
// 64-element dot products on the matrix unit with V_WMMA_F32_16X16X4_F32
// (A = per-lane products laid out as the 16x4 f32 A-matrix, B = ones, so
// D[m][n] = rowsum_m; lane 0 holds rows 0-7 of column 0, lane 16 rows 8-15).
// ---------------------------------------------------------------------------
typedef __attribute__((ext_vector_type(2))) float v2f;
typedef __attribute__((ext_vector_type(8))) float v8f;

__global__ void finalize_kernel(const unsigned int* __restrict__ partial,
                                int nblocks, float* __restrict__ out) {
    const int lane = threadIdx.x;            // 0..31, one full wave32
    // A-matrix 16x4 f32 layout: lane L<16 holds A[L][0],A[L][1];
    // lane L>=16 holds A[L-16][2],A[L-16][3]. Element c = 16*k + m.
    const int m  = lane & 15;
    const int kb = (lane < 16) ? 0 : 2;
    const int c0 = (kb + 0) * 16 + m;
    const int c1 = (kb + 1) * 16 + m;

    unsigned int h0 = 0u, h1 = 0u;
    for (int b = 0; b < nblocks; ++b) {
        h0 += partial[b * 64 + c0];
        h1 += partial[b * 64 + c1];
    }

    v2f aLoss, aCnt, bOnes;
    aLoss.x = (float)h0 * kTab.loss[c0];
    aLoss.y = (float)h1 * kTab.loss[c1];
    aCnt.x  = (float)h0 * (float)kTab.cnt[c0];
    aCnt.y  = (float)h1 * (float)kTab.cnt[c1];
    bOnes.x = 1.0f; bOnes.y = 1.0f;

    v8f z = {};
    // 8 args: (neg_a, A, neg_b, B, c_mod, C, reuse_a, reuse_b)
    v8f dLoss = __builtin_amdgcn_wmma_f32_16x16x4_f32(
        false, aLoss, false, bOnes, (short)0, z, false, false);
    v8f dCnt  = __builtin_amdgcn_wmma_f32_16x16x4_f32(
        false, aCnt,  false, bOnes, (short)0, z, false, false);

    float pl = 0.0f, pc = 0.0f;
    #pragma unroll
    for (int r = 0; r < 8; ++r) { pl += dLoss[r]; pc += dCnt[r]; }

    __shared__ float sl[2], sc[2];
    if (lane == 0)  { sl[0] = pl; sc[0] = pc; }
    if (lane == 16) { sl[1] = pl; sc[1] = pc; }
    __syncthreads();
    if (lane == 0) {
        float totalLoss = sl[0] + sl[1];
        float totalCnt  = sc[0] + sc[1];
        float denom = totalCnt < 1.0f ? 1.0f : totalCnt;
        out[0] = totalLoss / denom;
    }
}

// ---------------------------------------------------------------------------
// Launch: ignore d_in[0] (embeddings are unused by the reference — avoiding
// that 128 MB read is the key optimization; remaining traffic is ~0.5 MB).
// ---------------------------------------------------------------------------
extern "C" void kernel_launch(void* const* d_in, const int* in_sizes, int n_in,
                              void* d_out, int out_size, void* d_ws, size_t ws_size,
                              hipStream_t stream) {
    const int* idx = (const int*)d_in[1];
    int n = in_sizes[1];
    unsigned int* partial = (unsigned int*)d_ws;  // NBLK*64 u32 = 32 KB

    hist_kernel<<<NBLK, NTHR, 0, stream>>>(idx, n, partial);
    finalize_kernel<<<1, 32, 0, stream>>>(partial, NBLK, (float*)d_out);
}